// GatHIVNet_4398046511481
// MI455X (gfx1250) — compile-verified
//
#include <hip/hip_runtime.h>
#include <math.h>

#define NNODES 100000   // divisible by 16 -> no GEMM edge tiles needed
#define NEDGES 1600000
#define HID 128
#define NGRAPH 2000
#define BN_EPS 1e-5f
#define NEG_SLOPE 0.2f

typedef float v2f __attribute__((ext_vector_type(2)));
typedef float v8f __attribute__((ext_vector_type(8)));

// Order-preserving float <-> uint map for atomicMax on floats (incl. negatives).
__device__ __forceinline__ unsigned f2ord(float f) {
  unsigned b = __float_as_uint(f);
  return (b & 0x80000000u) ? ~b : (b | 0x80000000u);
}
__device__ __forceinline__ float ord2f(unsigned k) {
  unsigned b = (k & 0x80000000u) ? (k & 0x7FFFFFFFu) : ~k;
  return __uint_as_float(b);
}
// f2ord(-inf) == 0x007FFFFF
#define EMAX_NEG_INF 0x007FFFFFu

// ---------------- Atom encoder: h[n,c] = sum_f emb[f, x[n,f], c] ----------------
// N*HID is an exact multiple of 256 -> no tail guard.
__global__ __launch_bounds__(256) void atom_encode(const int* __restrict__ x,
    const float* __restrict__ emb, float* __restrict__ h) {
  int idx = blockIdx.x * 256 + threadIdx.x;
  int n = idx >> 7, c = idx & 127;
  float s = 0.f;
#pragma unroll
  for (int f = 0; f < 9; ++f) {
    int v = x[n * 9 + f];
    s += emb[(f * 119 + v) * HID + c];
  }
  h[idx] = s;
}

// ---------------- GEMM: O[N,128] = H[N,128] @ W[128,128] via V_WMMA_F32_16X16X4_F32
// One workgroup = 16 output rows; 8 waves each own a 16-column tile.
// NNODES % 16 == 0, so no bounds checks anywhere (straight-line stores).
__global__ __launch_bounds__(256) void gemm_h_wmma(const float* __restrict__ H,
    const float* __restrict__ W, float* __restrict__ O) {
  __shared__ float sH[16 * 132];  // stride 132 to dodge 16-way bank conflicts
  int row0 = blockIdx.x * 16;
  int tid = threadIdx.x;
  for (int i = tid; i < 16 * HID; i += 256) {
    int r = i >> 7, c = i & 127;
    sH[r * 132 + c] = H[(row0 + r) * HID + c];
  }
  __syncthreads();
  int wave = tid >> 5, lane = tid & 31;
  int col0 = wave * 16;
  int m = lane & 15;   // A row / B col / D col index for this lane
  int hi = lane >> 4;  // lane half selects K pair (A/B) and M half (D)
  v8f acc = {};
#pragma unroll
  for (int k = 0; k < HID; k += 4) {
    v2f a, b;
    // 16x4 f32 A layout: v0 = K=0(lo lanes)/K=2(hi lanes), v1 = K=1/K=3
    a.x = sH[m * 132 + k + 2 * hi + 0];
    a.y = sH[m * 132 + k + 2 * hi + 1];
    // 4x16 f32 B layout mirrors C/D striping: row K across 2 VGPRs, cols across lanes
    b.x = W[(k + 2 * hi + 0) * HID + col0 + m];
    b.y = W[(k + 2 * hi + 1) * HID + col0 + m];
    acc = __builtin_amdgcn_wmma_f32_16x16x4_f32(false, a, false, b, (short)0, acc,
                                                false, false);
  }
  // D layout: VGPR r -> M = r (lanes 0-15) / M = 8+r (lanes 16-31), N = lane&15
#pragma unroll
  for (int r = 0; r < 8; ++r) {
    O[(row0 + r + 8 * hi) * HID + col0 + m] = acc[r];
  }
}

// ---------------- Per-node attention logits ----------------
__global__ __launch_bounds__(256) void calc_alpha(const float* __restrict__ hp,
    const float* __restrict__ avs, const float* __restrict__ avd,
    float* __restrict__ asrc, float* __restrict__ adst, int heads) {
  int idx = blockIdx.x * 256 + threadIdx.x;
  if (idx >= NNODES * heads) return;
  int node = idx / heads, hd = idx - node * heads;
  int dh = HID / heads;
  const float* row = hp + node * HID + hd * dh;
  float ss = 0.f, sd = 0.f;
  for (int k = 0; k < dh; ++k) {
    float v = row[k];
    ss += v * avs[hd * dh + k];
    sd += v * avd[hd * dh + k];
  }
  asrc[idx] = ss;
  adst[idx] = sd;
}

// ---------------- Per-layer init: acc=bias, emax=-inf, denom=0, BN sums=0 --------
__global__ __launch_bounds__(256) void init_layer(float* __restrict__ hacc,
    const float* __restrict__ bias, unsigned* __restrict__ emax,
    float* __restrict__ denom, float* __restrict__ bnsum, float* __restrict__ bnsq) {
  int idx = blockIdx.x * 256 + threadIdx.x;
  hacc[idx] = bias[idx & 127];
  if (idx < NNODES * 8) { emax[idx] = EMAX_NEG_INF; denom[idx] = 0.f; }
  if (idx < HID) { bnsum[idx] = 0.f; bnsq[idx] = 0.f; }
}

__device__ __forceinline__ void edge_nodes(const int* __restrict__ ei, int e,
                                           int& s, int& d) {
  if (e < NEDGES) { s = ei[e]; d = ei[NEDGES + e]; }
  else { s = e - NEDGES; d = s; }  // implicit self-loop
}

// ---------------- Edge pass 1: segment max of leaky_relu logits ----------------
__global__ __launch_bounds__(256) void edge_max(const int* __restrict__ ei,
    const float* __restrict__ asrc, const float* __restrict__ adst,
    unsigned* __restrict__ emax, int heads) {
  int e = blockIdx.x * 256 + threadIdx.x;
  if (e >= NEDGES + NNODES) return;
  int s, d; edge_nodes(ei, e, s, d);
  for (int h = 0; h < heads; ++h) {
    float v = asrc[s * heads + h] + adst[d * heads + h];
    v = v > 0.f ? v : v * NEG_SLOPE;
    atomicMax(&emax[d * heads + h], f2ord(v));
  }
}

// ---------------- Edge pass 2: softmax denominator ----------------
__global__ __launch_bounds__(256) void edge_den(const int* __restrict__ ei,
    const float* __restrict__ asrc, const float* __restrict__ adst,
    const unsigned* __restrict__ emax, float* __restrict__ denom, int heads) {
  int e = blockIdx.x * 256 + threadIdx.x;
  if (e >= NEDGES + NNODES) return;
  int s, d; edge_nodes(ei, e, s, d);
  for (int h = 0; h < heads; ++h) {
    float v = asrc[s * heads + h] + adst[d * heads + h];
    v = v > 0.f ? v : v * NEG_SLOPE;
    float p = __expf(v - ord2f(emax[d * heads + h]));
    atomicAdd(&denom[d * heads + h], p);
  }
}

// ---------------- Edge pass 3: weighted scatter-add (1 wave per edge) ----------
__global__ __launch_bounds__(256) void edge_agg(const int* __restrict__ ei,
    const float* __restrict__ hp, const float* __restrict__ asrc,
    const float* __restrict__ adst, const unsigned* __restrict__ emax,
    const float* __restrict__ denom, float* __restrict__ out, int heads) {
  int idx = blockIdx.x * 256 + threadIdx.x;
  int e = idx >> 5, lane = idx & 31;
  if (e >= NEDGES + NNODES) return;
  int s, d; edge_nodes(ei, e, s, d);
  int c0 = lane * 4;              // 4 channels per lane; 4 | dh so single head
  int dh = HID / heads;
  int hd = c0 / dh;
  float v = asrc[s * heads + hd] + adst[d * heads + hd];
  v = v > 0.f ? v : v * NEG_SLOPE;
  float alpha = __expf(v - ord2f(emax[d * heads + hd])) / denom[d * heads + hd];
  const float4 hv = *(const float4*)(hp + s * HID + c0);
  float* o = out + d * HID + c0;
  atomicAdd(o + 0, hv.x * alpha);
  atomicAdd(o + 1, hv.y * alpha);
  atomicAdd(o + 2, hv.z * alpha);
  atomicAdd(o + 3, hv.w * alpha);
}

// ---------------- BatchNorm stats: coalesced per-channel partial sums ----------
__global__ __launch_bounds__(256) void bn_partial(const float* __restrict__ xin,
    float* __restrict__ bnsum, float* __restrict__ bnsq) {
  int c = threadIdx.x & 127;
  int r = blockIdx.x * 2 + (threadIdx.x >> 7);
  int rstep = gridDim.x * 2;
  float s = 0.f, s2 = 0.f;
  for (; r < NNODES; r += rstep) {
    float v = xin[r * HID + c];
    s += v; s2 += v * v;
  }
  atomicAdd(&bnsum[c], s);
  atomicAdd(&bnsq[c], s2);
}

// ---------------- BN + ReLU + residual (in place into h) ----------------
__global__ __launch_bounds__(256) void bn_apply(const float* __restrict__ hacc,
    const float* __restrict__ bnsum, const float* __restrict__ bnsq,
    const float* __restrict__ gamma, const float* __restrict__ beta,
    float* __restrict__ h) {
  int idx = blockIdx.x * 256 + threadIdx.x;
  int c = idx & 127;
  const float inv_n = 1.f / (float)NNODES;
  float mean = bnsum[c] * inv_n;
  float var = bnsq[c] * inv_n - mean * mean;   // biased variance
  float v = gamma[c] * (hacc[idx] - mean) * rsqrtf(var + BN_EPS) + beta[c];
  v = fmaxf(v, 0.f);
  h[idx] = h[idx] + v;  // residual
}

// ---------------- Global mean pool ----------------
__global__ __launch_bounds__(256) void pool_init(float* __restrict__ gsum,
                                                 float* __restrict__ gcnt) {
  int idx = blockIdx.x * 256 + threadIdx.x;
  if (idx < NGRAPH * HID) gsum[idx] = 0.f;
  if (idx < NGRAPH) gcnt[idx] = 0.f;
}

__global__ __launch_bounds__(256) void pool(const int* __restrict__ batch,
    const float* __restrict__ h, float* __restrict__ gsum, float* __restrict__ gcnt) {
  int idx = blockIdx.x * 256 + threadIdx.x;
  int n = idx >> 7, c = idx & 127;
  int g = batch[n];
  atomicAdd(&gsum[g * HID + c], h[idx]);
  if (c == 0) atomicAdd(&gcnt[g], 1.f);
}

// ---------------- MLP head: 128 -> 64 -> 32 -> 1, one block per graph ----------
__global__ __launch_bounds__(64) void mlp_head(const float* __restrict__ gsum,
    const float* __restrict__ gcnt, const float* __restrict__ w1,
    const float* __restrict__ b1, const float* __restrict__ w2,
    const float* __restrict__ b2, const float* __restrict__ w3,
    const float* __restrict__ b3, float* __restrict__ out) {
  int g = blockIdx.x;
  __shared__ float gv[HID];
  __shared__ float h1[64];
  __shared__ float h2[32];
  float cnt = fmaxf(gcnt[g], 1.f);
  for (int i = threadIdx.x; i < HID; i += 64) gv[i] = gsum[g * HID + i] / cnt;
  __syncthreads();
  int t = threadIdx.x;
  {
    float a = b1[t];
    for (int k = 0; k < HID; ++k) a += gv[k] * w1[k * 64 + t];
    h1[t] = fmaxf(a, 0.f);
  }
  __syncthreads();
  if (t < 32) {
    float a = b2[t];
    for (int k = 0; k < 64; ++k) a += h1[k] * w2[k * 32 + t];
    h2[t] = fmaxf(a, 0.f);
  }
  __syncthreads();
  if (t == 0) {
    float a = b3[0];
    for (int k = 0; k < 32; ++k) a += h2[k] * w3[k];
    out[g] = a;
  }
}

extern "C" void kernel_launch(void* const* d_in, const int* in_sizes, int n_in,
                              void* d_out, int out_size, void* d_ws, size_t ws_size,
                              hipStream_t stream) {
  (void)in_sizes; (void)n_in; (void)out_size; (void)ws_size;
  const int*   x            = (const int*)d_in[0];
  const int*   ei           = (const int*)d_in[1];
  const int*   batch        = (const int*)d_in[2];
  const float* atom_emb     = (const float*)d_in[3];
  const float* W_gat        = (const float*)d_in[4];
  const float* bias_gat     = (const float*)d_in[5];
  const float* att_src_a    = (const float*)d_in[6];
  const float* att_dst_a    = (const float*)d_in[7];
  const float* att_src_last = (const float*)d_in[8];
  const float* att_dst_last = (const float*)d_in[9];
  const float* bn_gamma     = (const float*)d_in[10];
  const float* bn_beta      = (const float*)d_in[11];
  const float* w1 = (const float*)d_in[12];
  const float* b1 = (const float*)d_in[13];
  const float* w2 = (const float*)d_in[14];
  const float* b2 = (const float*)d_in[15];
  const float* w3 = (const float*)d_in[16];
  const float* b3 = (const float*)d_in[17];
  float* out = (float*)d_out;

  // Carve workspace (deterministic layout, re-initialized every call).
  char* ws = (char*)d_ws;
  size_t off = 0;
  auto carve = [&](size_t bytes) -> void* {
    void* p = ws + off;
    off += (bytes + 255) & ~(size_t)255;
    return p;
  };
  float*    h     = (float*)carve((size_t)NNODES * HID * 4);
  float*    hp    = (float*)carve((size_t)NNODES * HID * 4);
  float*    hacc  = (float*)carve((size_t)NNODES * HID * 4);
  float*    asrc  = (float*)carve((size_t)NNODES * 8 * 4);
  float*    adst  = (float*)carve((size_t)NNODES * 8 * 4);
  unsigned* emax  = (unsigned*)carve((size_t)NNODES * 8 * 4);
  float*    denom = (float*)carve((size_t)NNODES * 8 * 4);
  float*    bnsum = (float*)carve(HID * 4);
  float*    bnsq  = (float*)carve(HID * 4);
  float*    gsum  = (float*)carve((size_t)NGRAPH * HID * 4);
  float*    gcnt  = (float*)carve((size_t)NGRAPH * 4);

  dim3 blk(256);
  const int gNC = (NNODES * HID) / 256;  // exact
  const int Etot = NEDGES + NNODES;

  atom_encode<<<gNC, blk, 0, stream>>>(x, atom_emb, h);

  for (int i = 0; i < 4; ++i) {
    int heads = (i < 3) ? 8 : 1;
    const float* W    = W_gat + (size_t)i * HID * HID;
    const float* bias = bias_gat + (size_t)i * HID;
    const float* avs  = (i < 3) ? att_src_a + (size_t)i * HID : att_src_last;
    const float* avd  = (i < 3) ? att_dst_a + (size_t)i * HID : att_dst_last;

    gemm_h_wmma<<<NNODES / 16, blk, 0, stream>>>(h, W, hp);
    calc_alpha<<<(NNODES * heads + 255) / 256, blk, 0, stream>>>(hp, avs, avd,
                                                                 asrc, adst, heads);
    init_layer<<<gNC, blk, 0, stream>>>(hacc, bias, emax, denom, bnsum, bnsq);
    edge_max<<<(Etot + 255) / 256, blk, 0, stream>>>(ei, asrc, adst, emax, heads);
    edge_den<<<(Etot + 255) / 256, blk, 0, stream>>>(ei, asrc, adst, emax, denom, heads);
    edge_agg<<<((size_t)Etot * 32 + 255) / 256, blk, 0, stream>>>(
        ei, hp, asrc, adst, emax, denom, hacc, heads);
    bn_partial<<<512, blk, 0, stream>>>(hacc, bnsum, bnsq);
    bn_apply<<<gNC, blk, 0, stream>>>(hacc, bnsum, bnsq, bn_gamma + (size_t)i * HID,
                                      bn_beta + (size_t)i * HID, h);
  }

  pool_init<<<(NGRAPH * HID) / 256, blk, 0, stream>>>(gsum, gcnt);
  pool<<<gNC, blk, 0, stream>>>(batch, h, gsum, gcnt);
  mlp_head<<<NGRAPH, dim3(64), 0, stream>>>(gsum, gcnt, w1, b1, w2, b2, w3, b3, out);
}